// EMLTree_27127013441561
// MI455X (gfx1250) — compile-verified
//
#include <hip/hip_runtime.h>
#include <math.h>

// ---------------------------------------------------------------------------
// EML tree on MI455X (gfx1250).
//   B = 32768 samples, 1024 leaves, 1023 internal nodes, 10 levels.
//   Leaf stage is a (B x 3) * (3 x 1024) fp32 contraction -> V_WMMA_F32_16X16X4_F32
//   (K=4, K3 padded with zero). Node math (gate + complex exp - complex log,
//   clamp to 1e300) is done in fp64 per the reference.
//
// d_out layout (as double*):
//   [0                 .. 2*B)                    : out  (complex128 of current[:,0])
//   [2*B               .. 2*B+3072)               : leaf_probs (1024 x 3, f64)
//   [2*B+3072          .. 2*B+3072+2046)          : gate_probs (1023 x 2, f64)
//   then eml_outputs[0..9], level l has B x (512>>l) complex128, row-major,
//   (re, im) interleaved.
// ---------------------------------------------------------------------------

typedef __attribute__((ext_vector_type(2))) float v2f;
typedef __attribute__((ext_vector_type(8))) float v8f;

#define EML_B        32768
#define EML_NLEAF    1024
#define EML_CLAMP_V  1e300
#define EML_THR      (1.0 - 2.220446049250313e-16)   // 1 - eps(f64)

__device__ __forceinline__ double eml_clampv(double v) {
    if (isnan(v)) return 0.0;
    if (v >  EML_CLAMP_V) return  EML_CLAMP_V;
    if (v < -EML_CLAMP_V) return -EML_CLAMP_V;
    return v;
}

// One EML node: left/right children (complex), gate sigmoids sl/sr.
__device__ __forceinline__ void eml_node(double Lr, double Li, double Rr, double Ri,
                                         double sl, double sr,
                                         double& outr, double& outi) {
    const double oml = 1.0 - sl;
    const double omr = 1.0 - sr;
    const bool bl = sl > EML_THR;
    const bool br = sr > EML_THR;
    const double lr = bl ? 1.0 : fma(oml, Lr, sl);
    const double li = bl ? 0.0 : oml * Li;
    const double rr = br ? 1.0 : fma(omr, Rr, sr);
    const double ri = br ? 0.0 : omr * Ri;
    // exp(left) : exp(lr) * (cos(li) + i sin(li))
    const double e = exp(lr);
    double sli, cli;
    sincos(li, &sli, &cli);
    const double er = e * cli;
    const double ei = e * sli;
    // log(right): 0.5*log(rr^2+ri^2) + i*atan2(ri, rr)
    const double lgr = 0.5 * log(fma(rr, rr, ri * ri));
    const double lgi = atan2(ri, rr);
    outr = eml_clampv(er - lgr);
    outi = eml_clampv(ei - lgi);
}

// ---------------------------------------------------------------------------
// Kernel 1: leaf softmax probs + gate sigmoids (fp64), written to output slots.
// ---------------------------------------------------------------------------
__global__ void eml_probs_kernel(const float* __restrict__ leaf_logits,
                                 const float* __restrict__ blend_logits,
                                 double* __restrict__ leaf_probs,
                                 double* __restrict__ gate_probs) {
    const int t = blockIdx.x * blockDim.x + threadIdx.x;
    if (t < EML_NLEAF) {
        const double a = (double)leaf_logits[t * 3 + 0];
        const double b = (double)leaf_logits[t * 3 + 1];
        const double c = (double)leaf_logits[t * 3 + 2];
        const double m = fmax(a, fmax(b, c));
        const double ea = exp(a - m), eb = exp(b - m), ec = exp(c - m);
        const double inv = 1.0 / (ea + eb + ec);
        leaf_probs[t * 3 + 0] = ea * inv;
        leaf_probs[t * 3 + 1] = eb * inv;
        leaf_probs[t * 3 + 2] = ec * inv;
    }
    if (t < (EML_NLEAF - 1) * 2) {
        gate_probs[t] = 1.0 / (1.0 + exp(-(double)blend_logits[t]));
    }
}

// ---------------------------------------------------------------------------
// Kernel 2: level 0 — WMMA leaf evaluation fused with the first EML level.
// One wave computes a 16(batch) x 16(leaf) tile with V_WMMA_F32_16X16X4_F32:
//   A = weights (16 leaves x 4: w0,w1,w2,0)   -> lanes 0-15: {w0,w1}, 16-31: {w2,0}
//   B = candidates (4 x 16 batch: 1,x,y,0)    -> lanes 0-15: {1,x},  16-31: {y,0}
//   D: lane = batch (mod 16); VGPR r = leaf row r (lanes<16) or 8+r (lanes>=16)
// Leaf pairs (2p,2p+1) therefore live in one lane -> fuse gate/exp/log locally.
// ---------------------------------------------------------------------------
__global__ __launch_bounds__(256) void eml_level0_wmma_kernel(
        const float* __restrict__ x,
        const float* __restrict__ y,
        const float* __restrict__ leaf_logits,
        const double* __restrict__ gate_probs,
        double* __restrict__ eml0) {
    const int lane   = threadIdx.x & 31;
    const int waveId = blockIdx.x * 8 + (threadIdx.x >> 5);
    const int tilesPerRow = EML_NLEAF / 16;             // 64 leaf tiles
    const int batchBase = (waveId / tilesPerRow) * 16;
    const int leafBase  = (waveId % tilesPerRow) * 16;
    const int half = lane >> 4;                          // 0: K0/K1, 1: K2/K3
    const int l15  = lane & 15;

    // A operand: softmax(leaf_logits) in f32, computed per lane for its leaf.
    const int leaf = leafBase + l15;
    const float a0 = leaf_logits[leaf * 3 + 0];
    const float a1 = leaf_logits[leaf * 3 + 1];
    const float a2 = leaf_logits[leaf * 3 + 2];
    const float mx = fmaxf(a0, fmaxf(a1, a2));
    const float e0 = __expf(a0 - mx), e1 = __expf(a1 - mx), e2 = __expf(a2 - mx);
    const float inv = 1.0f / (e0 + e1 + e2);
    v2f A;
    if (half == 0) { A.x = e0 * inv; A.y = e1 * inv; }
    else           { A.x = e2 * inv; A.y = 0.0f; }

    // B operand: candidate rows [1, x, y, 0] for this lane's batch column.
    const int batch = batchBase + l15;
    const float xv = x[batch];
    const float yv = y[batch];
    v2f Bv;
    if (half == 0) { Bv.x = 1.0f; Bv.y = xv; }
    else           { Bv.x = yv;   Bv.y = 0.0f; }

    v8f C = {};
    v8f D = __builtin_amdgcn_wmma_f32_16x16x4_f32(
        /*neg_a=*/false, A, /*neg_b=*/false, Bv,
        /*c_mod=*/(short)0, C, /*reuse_a=*/false, /*reuse_b=*/false);

    float dv[8];
#pragma unroll
    for (int i = 0; i < 8; ++i) dv[i] = D[i];

    // This lane holds 8 leaf values (rows half*8 .. half*8+7) for `batch`
    // -> 4 level-0 nodes. Global pair index = leafBase/2 + half*4 + p.
#pragma unroll
    for (int p = 0; p < 4; ++p) {
        const int g = (leafBase >> 1) + half * 4 + p;    // node / output column
        const double sl = gate_probs[g * 2 + 0];
        const double sr = gate_probs[g * 2 + 1];
        const double Lr = (double)dv[2 * p + 0];
        const double Rr = (double)dv[2 * p + 1];
        double zr, zi;
        eml_node(Lr, 0.0, Rr, 0.0, sl, sr, zr, zi);
        const long idx = ((long)batch * 512 + g) * 2;
        eml0[idx + 0] = zr;
        eml0[idx + 1] = zi;
    }
}

// ---------------------------------------------------------------------------
// Kernels 3..11: level l (>=1). One thread per (batch, pair). k varies fastest
// for coalesced 16B complex stores; prev reads are 32B-strided (L2-friendly).
// ---------------------------------------------------------------------------
__global__ __launch_bounds__(256) void eml_level_kernel(
        const double* __restrict__ gate_probs,
        const double* __restrict__ prev,   // B x (2*n_pairs) complex
        double* __restrict__ cur,          // B x n_pairs complex
        double* __restrict__ final_out,    // non-null on last level
        int n_pairs, int shift, int node_base) {
    const int tid = blockIdx.x * blockDim.x + threadIdx.x;
    const int k = tid & (n_pairs - 1);
    const int b = tid >> shift;
    if (b >= EML_B) return;

    const long pbase = ((long)b * (2 * n_pairs) + 2 * k) * 2;
    const double Lr = prev[pbase + 0];
    const double Li = prev[pbase + 1];
    const double Rr = prev[pbase + 2];
    const double Ri = prev[pbase + 3];
    const double sl = gate_probs[(node_base + k) * 2 + 0];
    const double sr = gate_probs[(node_base + k) * 2 + 1];

    double zr, zi;
    eml_node(Lr, Li, Rr, Ri, sl, sr, zr, zi);

    const long cidx = ((long)b * n_pairs + k) * 2;
    cur[cidx + 0] = zr;
    cur[cidx + 1] = zi;
    if (final_out != nullptr) {          // last level: n_pairs == 1, k == 0
        final_out[(long)b * 2 + 0] = zr;
        final_out[(long)b * 2 + 1] = zi;
    }
}

// ---------------------------------------------------------------------------
extern "C" void kernel_launch(void* const* d_in, const int* in_sizes, int n_in,
                              void* d_out, int out_size, void* d_ws, size_t ws_size,
                              hipStream_t stream) {
    (void)in_sizes; (void)n_in; (void)out_size; (void)d_ws; (void)ws_size;
    const float* x            = (const float*)d_in[0];
    const float* y            = (const float*)d_in[1];
    const float* leaf_logits  = (const float*)d_in[2];
    const float* blend_logits = (const float*)d_in[3];
    double* out = (double*)d_out;

    // Output offsets (in doubles).
    const long offOut   = 0;
    const long offLeafP = offOut + 2L * EML_B;            // 65536
    const long offGateP = offLeafP + 1024L * 3;           // +3072
    long emlOff[10];
    emlOff[0] = offGateP + 1023L * 2;                     // +2046
    for (int l = 1; l < 10; ++l)
        emlOff[l] = emlOff[l - 1] + 2L * EML_B * (512 >> (l - 1));

    double* leafP = out + offLeafP;
    double* gateP = out + offGateP;

    // 1) probs (2048 threads covers both 1024 leaves and 2046 gate entries)
    eml_probs_kernel<<<8, 256, 0, stream>>>(leaf_logits, blend_logits, leafP, gateP);

    // 2) level 0: WMMA leaf eval + first EML level.
    //    (B/16)*(1024/16) = 131072 wave tiles, 8 waves per block -> 16384 blocks.
    eml_level0_wmma_kernel<<<16384, 256, 0, stream>>>(x, y, leaf_logits, gateP,
                                                      out + emlOff[0]);

    // 3) levels 1..9
    for (int l = 1; l < 10; ++l) {
        const int n_pairs = 512 >> l;
        const int shift = 9 - l;                          // n_pairs == 1 << shift
        const int node_base = 1024 - (1024 >> l);
        const long total = (long)EML_B * n_pairs;
        const int blocks = (int)(total / 256);            // exact for all levels
        double* final_out = (l == 9) ? (out + offOut) : nullptr;
        eml_level_kernel<<<blocks, 256, 0, stream>>>(
            gateP, out + emlOff[l - 1], out + emlOff[l], final_out,
            n_pairs, shift, node_base);
    }
}